// MyLeNetMatStochBU_23940147708641
// MI455X (gfx1250) — compile-verified
//
#include <hip/hip_runtime.h>
#include <hip/hip_bf16.h>

typedef __attribute__((ext_vector_type(2))) float v2f;
typedef __attribute__((ext_vector_type(8))) float v8f;

// Generic stochastic-strided-conv layer as im2col GEMM on f32 WMMA.
//   out[b,o,y,x] = relu( (mask[y,x] ? sum_k W[o,k]*patch[k,(b,y,x)] : 0) + bias[o] )
// patch address: ph = selh[y,x] + y*stride (in-bounds by construction).
// Layer 4 (plain VALID conv): selh/selw/mask = null, stride = 1, no relu.
//
// LDS tiles use a K-pair (float2) layout so each WMMA fragment is one
// ds_load_b64; pair-row stride 80 float2s => rows k2 and k2+1 (read by the
// two half-waves in the same instruction) hit disjoint 32-bank halves.
__global__ __launch_bounds__(128)
void sconv_wmma_f32_kernel(const float* __restrict__ in,
                           const float* __restrict__ wgt,   // [Cout][Cin*9] row-major
                           const float* __restrict__ bias,  // [Cout]
                           const int*   __restrict__ selh,  // [OH][OW] or null
                           const int*   __restrict__ selw,  // [OH][OW] or null
                           const int*   __restrict__ mask,  // [OH][OW] or null
                           float* __restrict__ out,         // [B][Cout][OH][OW]
                           int Cin, int Cout, int H, int W,
                           int OH, int OW, int stride, int doRelu)
{
    const int Ksz  = Cin * 9;
    const int OHOW = OH * OW;
    const int HW   = H * W;

    __shared__ v2f A_pair[16][80];   // K-pairs x M(64), 10.0 KB
    __shared__ v2f B_pair[16][80];   // K-pairs x N(64), 10.0 KB
    __shared__ int koff_s[7200];     // im2col offset c*HW + r*W + s per k (max Cin=800)
    __shared__ int pbase_s[64];      // b*Cin*HW + ph*W + pw
    __shared__ int nbase_s[64];      // b*Cout*OHOW + y*OW + x
    __shared__ int mask_s[64];

    const int t      = threadIdx.x;
    const int lane   = t & 31;
    const int waveId = t >> 5;
    const int lane16 = lane & 15;
    const int hl     = lane >> 4;        // half-wave: 0 -> K{0,1}, 1 -> K{2,3}

    const int tileN = blockIdx.x * 64;   // position tile (always full: B=64)
    const int m0    = blockIdx.y * 64;   // Cout tile (may have tail)

    // Per-tile patch coordinates (once per block).
    if (t < 64) {
        int n   = tileN + t;
        int b   = n / OHOW;
        int rem = n - b * OHOW;
        int y   = rem / OW;
        int x   = rem - y * OW;
        int sh  = selh ? selh[rem] : 0;
        int sw  = selw ? selw[rem] : 0;
        int ph  = y * stride + sh;
        int pw  = x * stride + sw;
        pbase_s[t] = b * Cin * HW + ph * W + pw;
        nbase_s[t] = b * Cout * OHOW + rem;
        mask_s[t]  = mask ? mask[rem] : 1;
    }
    // im2col k -> input offset table (once per block; removes all divisions
    // from the per-chunk staging loops).
    for (int k = t; k < Ksz; k += 128) {
        int c   = k / 9;
        int rem = k - c * 9;
        int r   = rem / 3;
        int s   = rem - r * 3;
        koff_s[k] = c * HW + r * W + s;
    }
    __syncthreads();

    v8f acc0 = {}, acc1 = {}, acc2 = {}, acc3 = {};

    const int mloc = (waveId << 4) + lane16;   // this wave's row in the A tile
    const int numKChunks = (Ksz + 31) >> 5;

    for (int kc = 0; kc < numKChunks; ++kc) {
        const int k0 = kc << 5;

        // Stage A: 16 K-pairs x 64 M. Lanes walk consecutive K within a
        // weight row -> coalesced global reads; weights are L2-resident.
        #pragma unroll
        for (int i = 0; i < 8; ++i) {
            int e  = t + i * 128;
            int k2 = e & 15;
            int m  = e >> 4;
            int o  = m0 + m;
            int kg = k0 + (k2 << 1);
            float v0 = 0.0f, v1 = 0.0f;
            if (o < Cout) {
                const float* wp = wgt + o * Ksz + kg;
                if (kg     < Ksz) v0 = wp[0];
                if (kg + 1 < Ksz) v1 = wp[1];
            }
            v2f av; av.x = v0; av.y = v1;
            A_pair[k2][m] = av;
        }
        // Stage B: gathered im2col patches, 16 K-pairs x 64 N.
        // koff_s reads are wave-uniform broadcasts; 64-bit LDS stores with
        // consecutive n across lanes are conflict-free.
        #pragma unroll
        for (int i = 0; i < 8; ++i) {
            int e  = t + i * 128;
            int k2 = e >> 6;
            int n  = e & 63;
            int kg = k0 + (k2 << 1);
            int pb = pbase_s[n];
            float v0 = 0.0f, v1 = 0.0f;
            if (kg     < Ksz) v0 = in[pb + koff_s[kg]];
            if (kg + 1 < Ksz) v1 = in[pb + koff_s[kg + 1]];
            v2f bv; bv.x = v0; bv.y = v1;
            B_pair[k2][n] = bv;
        }
        __syncthreads();

        // 8 K-steps (of 4) x 4 N-subtiles = 32 WMMAs per chunk per wave.
        // One ds_load_b64 per fragment: row = kk/2 for lanes<16, +1 for >=16.
        #pragma unroll
        for (int kk2 = 0; kk2 < 16; kk2 += 2) {
            int row = kk2 + hl;
            v2f a  = A_pair[row][mloc];
            v2f b0 = B_pair[row][lane16];
            v2f b1 = B_pair[row][16 + lane16];
            v2f b2 = B_pair[row][32 + lane16];
            v2f b3 = B_pair[row][48 + lane16];
            acc0 = __builtin_amdgcn_wmma_f32_16x16x4_f32(false, a, false, b0, (short)0, acc0, false, false);
            acc1 = __builtin_amdgcn_wmma_f32_16x16x4_f32(false, a, false, b1, (short)0, acc1, false, false);
            acc2 = __builtin_amdgcn_wmma_f32_16x16x4_f32(false, a, false, b2, (short)0, acc2, false, false);
            acc3 = __builtin_amdgcn_wmma_f32_16x16x4_f32(false, a, false, b3, (short)0, acc3, false, false);
        }
        __syncthreads();
    }

    // Epilogue: D layout — lane<16: M=r, lane>=16: M=r+8; N = lane16.
    #pragma unroll
    for (int r = 0; r < 8; ++r) {
        int o = m0 + (waveId << 4) + (hl ? r + 8 : r);
        if (o < Cout) {
            float bv = bias[o];
            float vals[4] = { acc0[r], acc1[r], acc2[r], acc3[r] };
            #pragma unroll
            for (int nb = 0; nb < 4; ++nb) {
                int nl = (nb << 4) + lane16;
                float v = mask_s[nl] ? vals[nb] : 0.0f;   // masked cells: never computed
                v += bv;                                   // bias added everywhere
                if (doRelu) v = fmaxf(v, 0.0f);
                out[nbase_s[nl] + o * OHOW] = v;
            }
        }
    }
}

extern "C" void kernel_launch(void* const* d_in, const int* in_sizes, int n_in,
                              void* d_out, int out_size, void* d_ws, size_t ws_size,
                              hipStream_t stream) {
    (void)in_sizes; (void)n_in; (void)out_size; (void)ws_size;

    const float* x  = (const float*)d_in[0];
    const float* w1 = (const float*)d_in[1];
    const float* b1 = (const float*)d_in[2];
    const float* w2 = (const float*)d_in[3];
    const float* b2 = (const float*)d_in[4];
    const float* w3 = (const float*)d_in[5];
    const float* b3 = (const float*)d_in[6];
    const float* w4 = (const float*)d_in[7];
    const float* b4 = (const float*)d_in[8];
    const int* selh1 = (const int*)d_in[9];
    const int* selw1 = (const int*)d_in[10];
    const int* mask1 = (const int*)d_in[11];
    const int* selh2 = (const int*)d_in[12];
    const int* selw2 = (const int*)d_in[13];
    const int* mask2 = (const int*)d_in[14];
    const int* selh3 = (const int*)d_in[15];
    const int* selw3 = (const int*)d_in[16];
    const int* mask3 = (const int*)d_in[17];

    float* ws = (float*)d_ws;
    // act1: 64*200*65*65 = 54,080,000 floats at offset 0
    // act2: 64*400*31*31 = 24,601,600 floats after act1
    // act3: 64*800*15*15 = 11,520,000 floats, reuses act1's space (act1 dead)
    float* act1 = ws;
    float* act2 = ws + 54080000;
    float* act3 = ws;

    dim3 blk(128);

    // L1: (64,3,132,132) -> (64,200,65,65); N = 270400 -> 4225 tiles, K = 27
    {
        dim3 g(270400 / 64, (200 + 63) / 64);
        sconv_wmma_f32_kernel<<<g, blk, 0, stream>>>(
            x, w1, b1, selh1, selw1, mask1, act1,
            3, 200, 132, 132, 65, 65, 2, 1);
    }
    // L2: -> (64,400,31,31); N = 61504 -> 961 tiles, K = 1800
    {
        dim3 g(61504 / 64, (400 + 63) / 64);
        sconv_wmma_f32_kernel<<<g, blk, 0, stream>>>(
            act1, w2, b2, selh2, selw2, mask2, act2,
            200, 400, 65, 65, 31, 31, 2, 1);
    }
    // L3: -> (64,800,15,15); N = 14400 -> 225 tiles, K = 3600
    {
        dim3 g(14400 / 64, (800 + 63) / 64);
        sconv_wmma_f32_kernel<<<g, blk, 0, stream>>>(
            act2, w3, b3, selh3, selw3, mask3, act3,
            400, 800, 31, 31, 15, 15, 2, 1);
    }
    // L4: plain VALID conv -> (64,10,13,13) flattened into d_out; K = 7200
    {
        dim3 g(10816 / 64, 1);
        sconv_wmma_f32_kernel<<<g, blk, 0, stream>>>(
            act3, w4, b4, nullptr, nullptr, nullptr, (float*)d_out,
            800, 10, 15, 15, 13, 13, 1, 0);
    }
}